// GCN_53755810677007
// MI455X (gfx1250) — compile-verified
//
#include <hip/hip_runtime.h>

typedef __attribute__((ext_vector_type(2))) float v2f;
typedef __attribute__((ext_vector_type(8))) float v8f;

#define NN 50000
#define NE 800000
#define FD 128

// ---------------- degree / normalization ----------------
__global__ void k_zero(float* __restrict__ p, int n) {
  int i = blockIdx.x * blockDim.x + threadIdx.x;
  if (i < n) p[i] = 0.0f;
}

__global__ void k_deg(const int* __restrict__ col, float* __restrict__ deg) {
  int e = blockIdx.x * blockDim.x + threadIdx.x;
  if (e < NE) atomicAdd(&deg[col[e]], 1.0f);
}

__global__ void k_dinv(const float* __restrict__ deg, float* __restrict__ dinv) {
  int i = blockIdx.x * blockDim.x + threadIdx.x;
  if (i < NN) {
    float d = deg[i] + 1.0f;  // +1 self loop -> always > 0
    dinv[i] = rsqrtf(d);
  }
}

// ---------------- dense GEMM via V_WMMA_F32_16X16X4_F32 ----------------
// grid.x = M/16 (row tiles), 8 waves/block each owning one 16-col tile (8*16=128).
__global__ void __launch_bounds__(256) k_gemm_wmma(const float* __restrict__ A,
                                                   const float* __restrict__ W,
                                                   float* __restrict__ Out) {
  const int lane = threadIdx.x & 31;
  const int wave = threadIdx.x >> 5;              // N-tile 0..7
  const int m    = (blockIdx.x << 4) + (lane & 15);
  const int n    = (wave << 4) + (lane & 15);
  const int kofs = (lane >> 4) << 1;              // 0 (lanes 0-15) or 2 (lanes 16-31)

  const float* ap = A + (size_t)m * FD + kofs;
  const float* bp = W + (size_t)kofs * FD + n;

  v8f acc = {};
  #pragma unroll 4
  for (int k = 0; k < FD; k += 4) {
    v2f a = *(const v2f*)(ap + k);                // A[m, k+kofs .. k+kofs+1]
    v2f b;
    b.x = bp[(size_t)k * FD];                     // W[k+kofs,   n]
    b.y = bp[(size_t)(k + 1) * FD];               // W[k+kofs+1, n]
    acc = __builtin_amdgcn_wmma_f32_16x16x4_f32(
        /*neg_a=*/false, a, /*neg_b=*/false, b,
        /*c_mod=*/(short)0, acc, /*reuse_a=*/false, /*reuse_b=*/false);
  }

  // C/D layout: VGPR v -> row v (lanes 0-15) / row v+8 (lanes 16-31)
  const int row0 = (blockIdx.x << 4) + ((lane >> 4) << 3);
  float* op = Out + (size_t)row0 * FD + n;
  #pragma unroll
  for (int v = 0; v < 8; ++v) op[(size_t)v * FD] = acc[v];
}

// ---------------- self-loop init: G[i,:] = T[i,:] * dinv[i]^2 ----------------
__global__ void k_selfloop(const float* __restrict__ T, const float* __restrict__ dinv,
                           float* __restrict__ G) {
  int idx = blockIdx.x * blockDim.x + threadIdx.x;   // NN*32 threads, float4 each
  if (idx >= NN * (FD / 4)) return;
  int node = idx >> 5;
  int fg   = (idx & 31) << 2;
  float di = dinv[node];
  float s  = di * di;
  float4 v = *(const float4*)(T + (size_t)node * FD + fg);
  float4 o = make_float4(v.x * s, v.y * s, v.z * s, v.w * s);
  *(float4*)(G + (size_t)node * FD + fg) = o;
}

// ---------------- edge scatter: wave per edge, 512B coalesced gather ----------------
__global__ void k_scatter(const float* __restrict__ T, const int* __restrict__ row,
                          const int* __restrict__ col, const float* __restrict__ dinv,
                          float* __restrict__ G) {
  long long idx = (long long)blockIdx.x * blockDim.x + threadIdx.x;
  int e = (int)(idx >> 5);
  if (e >= NE) return;
  int fg = ((int)idx & 31) << 2;
  int r = row[e], c = col[e];
  float nrm = dinv[r] * dinv[c];
  float4 v = *(const float4*)(T + (size_t)r * FD + fg);
  float* g = G + (size_t)c * FD + fg;
  atomicAdd(g + 0, v.x * nrm);   // non-returning global_atomic_add_f32, L2-resident
  atomicAdd(g + 1, v.y * nrm);
  atomicAdd(g + 2, v.z * nrm);
  atomicAdd(g + 3, v.w * nrm);
}

// ---------------- bias + relu epilogue ----------------
__global__ void k_bias_relu(const float* __restrict__ G, const float* __restrict__ bias,
                            float* __restrict__ H) {
  int idx = blockIdx.x * blockDim.x + threadIdx.x;
  if (idx >= NN * (FD / 4)) return;
  int node = idx >> 5;
  int fg   = (idx & 31) << 2;
  float4 v = *(const float4*)(G + (size_t)node * FD + fg);
  float4 b = *(const float4*)(bias + fg);
  float4 o = make_float4(fmaxf(v.x + b.x, 0.0f), fmaxf(v.y + b.y, 0.0f),
                         fmaxf(v.z + b.z, 0.0f), fmaxf(v.w + b.w, 0.0f));
  *(float4*)(H + (size_t)node * FD + fg) = o;
}

// ---------------- layer 3: GEMV (128 -> 1), wave per node ----------------
__global__ void __launch_bounds__(256) k_gemv(const float* __restrict__ H,
                                              const float* __restrict__ W3,
                                              float* __restrict__ z) {
  int lane = threadIdx.x & 31;
  int node = blockIdx.x * (blockDim.x >> 5) + (threadIdx.x >> 5);
  if (node >= NN) return;
  float4 h = *(const float4*)(H + (size_t)node * FD + (lane << 2));
  float4 w = *(const float4*)(W3 + (lane << 2));
  float s = h.x * w.x + h.y * w.y + h.z * w.z + h.w * w.w;
  #pragma unroll
  for (int off = 16; off > 0; off >>= 1) s += __shfl_xor(s, off, 32);
  if (lane == 0) z[node] = s;
}

__global__ void k_init3(const float* __restrict__ z, const float* __restrict__ dinv,
                        const float* __restrict__ b3, float* __restrict__ out) {
  int i = blockIdx.x * blockDim.x + threadIdx.x;
  if (i < NN) {
    float di = dinv[i];
    out[i] = z[i] * di * di + b3[0];
  }
}

__global__ void k_scatter3(const float* __restrict__ z, const int* __restrict__ row,
                           const int* __restrict__ col, const float* __restrict__ dinv,
                           float* __restrict__ out) {
  int e = blockIdx.x * blockDim.x + threadIdx.x;
  if (e < NE) {
    int r = row[e], c = col[e];
    atomicAdd(&out[c], z[r] * dinv[r] * dinv[c]);
  }
}

// ---------------- driver ----------------
extern "C" void kernel_launch(void* const* d_in, const int* in_sizes, int n_in,
                              void* d_out, int out_size, void* d_ws, size_t ws_size,
                              hipStream_t stream) {
  const float* x   = (const float*)d_in[0];
  const int*   ei  = (const int*)d_in[1];      // [2, NE]
  const float* W1  = (const float*)d_in[2];
  const float* b1  = (const float*)d_in[3];
  const float* W2  = (const float*)d_in[4];
  const float* b2  = (const float*)d_in[5];
  const float* W3  = (const float*)d_in[6];
  const float* b3  = (const float*)d_in[7];
  float* out = (float*)d_out;

  const int* row = ei;
  const int* col = ei + NE;

  // workspace layout (floats)
  const size_t NP = 50048;                     // NN padded to 64
  float* ws   = (float*)d_ws;
  float* dinv = ws;                            // NP
  float* z    = ws + NP;                       // NP (deg, then gemv result)
  float* T    = ws + 2 * NP;                   // NN*FD  gemm output
  float* G    = T + (size_t)NN * FD;           // NN*FD  aggregation accumulator
  float* H    = G + (size_t)NN * FD;           // NN*FD  post-relu features
  (void)ws_size; (void)n_in; (void)in_sizes; (void)out_size;

  const int B = 256;
  const int gN   = (NN + B - 1) / B;           // node-wise
  const int gE   = (NE + B - 1) / B;           // edge-wise
  const int gNF4 = (NN * (FD / 4) + B - 1) / B;      // 6250
  const int gEF  = (int)(((long long)NE * 32 + B - 1) / B); // 100000
  const int gM   = NN / 16;                    // 3125 row tiles
  const int gW   = (NN + 7) / 8;               // 6250 (wave per node)

  // degrees -> dinv (shared by all 3 layers)
  k_zero<<<gN, B, 0, stream>>>(z, NN);
  k_deg<<<gE, B, 0, stream>>>(col, z);
  k_dinv<<<gN, B, 0, stream>>>(z, dinv);

  // layer 1
  k_gemm_wmma<<<gM, B, 0, stream>>>(x, W1, T);
  k_selfloop<<<gNF4, B, 0, stream>>>(T, dinv, G);
  k_scatter<<<gEF, B, 0, stream>>>(T, row, col, dinv, G);
  k_bias_relu<<<gNF4, B, 0, stream>>>(G, b1, H);

  // layer 2
  k_gemm_wmma<<<gM, B, 0, stream>>>(H, W2, T);
  k_selfloop<<<gNF4, B, 0, stream>>>(T, dinv, G);
  k_scatter<<<gEF, B, 0, stream>>>(T, row, col, dinv, G);
  k_bias_relu<<<gNF4, B, 0, stream>>>(G, b2, H);

  // layer 3 (128 -> 1)
  k_gemv<<<gW, B, 0, stream>>>(H, W3, z);
  k_init3<<<gN, B, 0, stream>>>(z, dinv, b3, out);
  k_scatter3<<<gE, B, 0, stream>>>(z, row, col, dinv, out);
}